// GCNDropEdge_21921513079347
// MI455X (gfx1250) — compile-verified
//
#include <hip/hip_runtime.h>

typedef float v2f __attribute__((ext_vector_type(2)));
typedef float v8f __attribute__((ext_vector_type(8)));

#define NNODES 50000
#define NEDGES 800000
#define KMAX   128
#define BPAD   (KMAX + 2)   // LDS row stride (dwords): even => 8B-aligned pairs

// ---------------- utility kernels ----------------

__global__ void zero_kernel(float* __restrict__ p, int n) {
    int i = blockIdx.x * blockDim.x + threadIdx.x;
    if (i < n) p[i] = 0.0f;
}

__global__ void degree_kernel(const int* __restrict__ dst, float* __restrict__ deg, int E) {
    int e = blockIdx.x * blockDim.x + threadIdx.x;
    if (e < E) atomicAdd(&deg[dst[e]], 1.0f);
}

__global__ void invdeg_kernel(float* __restrict__ deg, int n) {
    int i = blockIdx.x * blockDim.x + threadIdx.x;
    if (i < n) deg[i] = 1.0f / fmaxf(deg[i], 1.0f);
}

// ---------------- dense GEMM via fp32 WMMA + LDS-staged B ----------------
// Out[M x N] = Xin[M x K] @ W[K x N], row-major.
// grid.x: groups of 8 M-tiles (one per wave), grid.y: N-tile (shared by block).
// The 16-wide B stripe is staged transposed in LDS ([16][K+2]) so the inner
// loop is guard-free: global_load_b64 (A pair) + ds_load_b64 (B pair) + wmma.
__global__ __launch_bounds__(256) void gemm_wmma_kernel(
    const float* __restrict__ Xin, const float* __restrict__ W,
    float* __restrict__ Out, int K, int N, int Mtiles)
{
    __shared__ float smem[16 * BPAD];

    int tid = threadIdx.x;
    int n0  = blockIdx.y << 4;

    // Stage B stripe: smem[n][k] = W[k][n0+n], zero-padded for ragged N.
    for (int idx = tid; idx < (K << 4); idx += 256) {
        int k = idx >> 4;
        int n = idx & 15;
        int col = n0 + n;
        smem[n * BPAD + k] = (col < N) ? W[(size_t)k * N + col] : 0.0f;
    }
    __syncthreads();

    int tM = (blockIdx.x << 3) + (tid >> 5);
    if (tM >= Mtiles) return;            // wave-uniform exit

    int lane = tid & 31;
    int half = lane >> 4;                // 0: lanes 0-15, 1: lanes 16-31
    int lid  = lane & 15;
    int m0   = tM << 4;

    const float* arow = Xin + (size_t)(m0 + lid) * K;   // A row (M = m0+lid)
    const float* brow = smem + lid * BPAD;              // B column (N = n0+lid)
    int kofs = half << 1;                // lane-half K offset per ISA layout

    v8f acc = {};
    for (int k0 = 0; k0 < K; k0 += 4) {
        int ka = k0 + kofs;
        v2f a = *(const v2f*)(arow + ka);   // global_load_b64 (8B aligned)
        v2f b = *(const v2f*)(brow + ka);   // ds_load_b64     (8B aligned)
        acc = __builtin_amdgcn_wmma_f32_16x16x4_f32(
            /*neg_a=*/false, a, /*neg_b=*/false, b,
            /*c_mod=*/(short)0, acc, /*reuse_a=*/false, /*reuse_b=*/false);
    }

    int ncol = n0 + lid;
    if (ncol < N) {                      // only diverges on the N=40 tail tile
        // D layout: VGPR r -> row m0 + r + 8*half, col ncol
        float* orow = Out + (size_t)(m0 + (half << 3)) * N + ncol;
#pragma unroll
        for (int r = 0; r < 8; ++r)
            orow[(size_t)r * N] = acc[r];
    }
}

// ---------------- edge gather + atomic scatter-add (SpMM) ----------------
// grid.x covers edges, grid.y covers float4 feature chunks (F/4 of them).
__global__ void scatter_kernel(const float* __restrict__ P,
                               const int* __restrict__ src,
                               const int* __restrict__ dst,
                               float* __restrict__ agg, int E, int F)
{
    int e = blockIdx.x * blockDim.x + threadIdx.x;
    if (e >= E) return;
    int c = blockIdx.y << 2;             // starting feature of this chunk
    int s = src[e];
    int d = dst[e];
    float4 v = *(const float4*)(P + (size_t)s * F + c);
    float* base = agg + (size_t)d * F + c;
    atomicAdd(base + 0, v.x);
    atomicAdd(base + 1, v.y);
    atomicAdd(base + 2, v.z);
    atomicAdd(base + 3, v.w);
}

// ---------------- normalize + bias (+ReLU) ----------------
__global__ void pointwise_kernel(const float* __restrict__ agg,
                                 const float* __restrict__ inv_deg,
                                 const float* __restrict__ bias,
                                 float* __restrict__ out, int F, int relu)
{
    int i = blockIdx.x * blockDim.x + threadIdx.x;
    int total = NNODES * F;
    if (i >= total) return;
    int node = i / F;
    int f    = i - node * F;
    float v = agg[i] * inv_deg[node] + bias[f];
    out[i] = relu ? fmaxf(v, 0.0f) : v;
}

// ---------------- launch sequence ----------------

extern "C" void kernel_launch(void* const* d_in, const int* in_sizes, int n_in,
                              void* d_out, int out_size, void* d_ws, size_t ws_size,
                              hipStream_t stream)
{
    const float* features = (const float*)d_in[0];
    const int*   src      = (const int*)  d_in[1];
    const int*   dst      = (const int*)  d_in[2];
    const float* W0       = (const float*)d_in[3];
    const float* b0       = (const float*)d_in[4];
    const float* W1       = (const float*)d_in[5];
    const float* b1       = (const float*)d_in[6];
    const float* W2       = (const float*)d_in[7];
    const float* b2       = (const float*)d_in[8];
    float*       out      = (float*)d_out;

    float* ws  = (float*)d_ws;
    float* deg = ws;                         // 50000 floats (becomes inv_deg)
    float* P   = ws + 50048;                 // projection buffer, 50000*64
    float* A   = P  + (size_t)NNODES * 64;   // aggregation buffer, 50000*64
    float* H   = A  + (size_t)NNODES * 64;   // hidden activations, 50000*64

    const int Mtiles     = NNODES / 16;            // 3125, exact
    const int gemmBlksX  = (Mtiles + 7) / 8;       // 391
    const int edgeBlocks = (NEDGES + 255) / 256;   // 3125, exact

    // --- degrees: zero, accumulate, invert (in place) ---
    zero_kernel  <<<(NNODES + 255) / 256, 256, 0, stream>>>(deg, NNODES);
    degree_kernel<<<edgeBlocks, 256, 0, stream>>>(dst, deg, NEDGES);
    invdeg_kernel<<<(NNODES + 255) / 256, 256, 0, stream>>>(deg, NNODES);

    // --- layer 0: 128 -> 64, ReLU ---
    {
        int F = 64, tot = NNODES * F;
        gemm_wmma_kernel<<<dim3(gemmBlksX, 4), 256, 0, stream>>>(features, W0, P, 128, F, Mtiles);
        zero_kernel     <<<(tot + 255) / 256, 256, 0, stream>>>(A, tot);
        scatter_kernel  <<<dim3(edgeBlocks, F / 4), 256, 0, stream>>>(P, src, dst, A, NEDGES, F);
        pointwise_kernel<<<(tot + 255) / 256, 256, 0, stream>>>(A, deg, b0, H, F, 1);
    }

    // --- layer 1: 64 -> 64, ReLU (H consumed by GEMM, then rewritten) ---
    {
        int F = 64, tot = NNODES * F;
        gemm_wmma_kernel<<<dim3(gemmBlksX, 4), 256, 0, stream>>>(H, W1, P, 64, F, Mtiles);
        zero_kernel     <<<(tot + 255) / 256, 256, 0, stream>>>(A, tot);
        scatter_kernel  <<<dim3(edgeBlocks, F / 4), 256, 0, stream>>>(P, src, dst, A, NEDGES, F);
        pointwise_kernel<<<(tot + 255) / 256, 256, 0, stream>>>(A, deg, b1, H, F, 1);
    }

    // --- layer 2: 64 -> 40, no activation, write d_out ---
    {
        int F = 40, tot = NNODES * F;   // ragged N handled by zero-padded LDS stripe
        gemm_wmma_kernel<<<dim3(gemmBlksX, 3), 256, 0, stream>>>(H, W2, P, 64, F, Mtiles);
        zero_kernel     <<<(tot + 255) / 256, 256, 0, stream>>>(A, tot);
        scatter_kernel  <<<dim3(edgeBlocks, F / 4), 256, 0, stream>>>(P, src, dst, A, NEDGES, F);
        pointwise_kernel<<<(tot + 255) / 256, 256, 0, stream>>>(A, deg, b2, out, F, 0);
    }
}